// AutoregressiveRAM_74483322847756
// MI455X (gfx1250) — compile-verified
//
#include <hip/hip_runtime.h>
#include <stdint.h>
#include <stddef.h>

// ---------------------------------------------------------------------------
// AutoregressiveRAM on MI455X (gfx1250)
//
// The recurrence is strictly sequential (4096 steps) -> one persistent
// 1024-thread workgroup on a single WGP; minimize per-step latency:
//   * state          : 8196 bits packed into 257 dwords, PING-PONG buffered in
//                      LDS -> only ONE __syncthreads per step
//   * connections    : packed two-per-dword, transposed [5][8192] = 160 KB,
//                      DMA'd into LDS once via the Tensor Data Mover
//                      (tensor_load_to_lds + s_wait_tensorcnt); per step each
//                      neuron needs 5 conflict-free ds_load_b32 (lanes read
//                      consecutive dwords -> 32 distinct banks, no conflicts)
//   * transition RAM : bit-packed to 1 MB, L2-resident (global_prefetch warm),
//                      8 independent random L2 loads in flight per thread
//   * new state      : wave32 __ballot -> one ds_store_b32 per wave per group
//   * output         : non-temporal stores (134 MB stream must not evict the
//                      1 MB RAM table from L2)
// WMMA is deliberately NOT used: the dense-matvec reformulation of the bit
// gather would move ~275 GB through L2 vs ~134 MB total for the direct form.
// ---------------------------------------------------------------------------

#define BITS        8192
#define NB_T        10
#define NB_P        (NB_T / 2)         // connection pairs per neuron
#define NB_I        4
#define TM_WORDS    32                 // 1024 RAM entries -> 32 packed dwords
#define THREADS     1024
#define NPT         (BITS / THREADS)   // 8 neurons per thread
#define ST_STRIDE   264                // dwords per state buffer (257 used)

// workspace layout (bytes)
#define WS_CONN_OFF 0                          // u32 [NB_P][BITS]  = 163840
#define WS_PI_OFF   163840                     // u16 [BITS]        = 16384
#define WS_PT_OFF   (163840 + 16384)           // u32 [BITS*32]     = 1 MB

#define SMEM_BYTES  (NB_P * BITS * 4 + 2 * ST_STRIDE * 4)   // 160KB + 2 state bufs

typedef uint32_t u32x4 __attribute__((ext_vector_type(4)));
typedef int      i32x8 __attribute__((ext_vector_type(8)));
typedef int      i32x4 __attribute__((ext_vector_type(4)));

// ---------------------------- pack kernels ---------------------------------

__global__ void pack_transition_kernel(const float* __restrict__ tm,
                                       uint32_t* __restrict__ pt) {
    int id = blockIdx.x * blockDim.x + threadIdx.x;      // n*32 + w
    if (id >= BITS * TM_WORDS) return;
    const float* src = tm + (size_t)id * 32;             // n*1024 + w*32
    uint32_t word = 0;
#pragma unroll
    for (int b = 0; b < 32; ++b)
        word |= (src[b] > 0.5f ? 1u : 0u) << b;
    pt[id] = word;
}

__global__ void pack_initial_kernel(const float* __restrict__ im,
                                    uint16_t* __restrict__ pi) {
    int n = blockIdx.x * blockDim.x + threadIdx.x;
    if (n >= BITS) return;
    uint32_t word = 0;
#pragma unroll
    for (int b = 0; b < (1 << NB_I); ++b)
        word |= (im[n * (1 << NB_I) + b] > 0.5f ? 1u : 0u) << b;
    pi[n] = (uint16_t)word;
}

// pack conn pairs: connP[p*BITS + n] = conn[n][2p] | conn[n][2p+1] << 16
__global__ void pack_conn_kernel(const int* __restrict__ conn,
                                 uint32_t* __restrict__ cp) {
    int id = blockIdx.x * blockDim.x + threadIdx.x;      // n*NB_P + p
    if (id >= BITS * NB_P) return;
    int n = id / NB_P, p = id % NB_P;
    uint32_t c0 = (uint32_t)conn[n * NB_T + 2 * p];      // address-MSB side
    uint32_t c1 = (uint32_t)conn[n * NB_T + 2 * p + 1];
    cp[p * BITS + n] = (c0 & 0xFFFFu) | (c1 << 16);
}

// ------------------------- sequential kernel -------------------------------

__global__ __launch_bounds__(THREADS, 1)
void autoreg_ram_seq_kernel(const uint32_t* __restrict__ connP_g,
                            const uint32_t* __restrict__ packedT,
                            const uint16_t* __restrict__ packedI,
                            const int*      __restrict__ init_conn,
                            float*          __restrict__ out,
                            int length) {
    extern __shared__ uint8_t smem[];
    uint32_t* connP = (uint32_t*)smem;                          // [NB_P][BITS]
    uint32_t* sbuf  = (uint32_t*)(smem + NB_P * BITS * 4);      // [2][ST_STRIDE]

    const int tid  = threadIdx.x;
    const int lane = tid & 31;
    const int wave = tid >> 5;

    // ---- TDM: DMA the 160 KB packed connection table into LDS (wave 0) ----
    if (wave == 0) {
        uint64_t ga = (uint64_t)(uintptr_t)connP_g;
        const uint32_t NELEM = (uint32_t)(NB_P * BITS);        // 40960 x 4B elements

        u32x4 g0;
        g0[0] = 1u;                                            // count=1, user mode
        g0[1] = 0u;                                            // lds_addr (bytes) = 0
        g0[2] = (uint32_t)ga;                                  // global_addr[31:0]
        g0[3] = (uint32_t)((ga >> 32) & 0x1FFFFFFu) | (2u << 30); // addr[56:32] | type=2

        i32x8 g1;
        g1[0] = (int)(2u << 16);                               // data_size = 4B
        g1[1] = (int)((NELEM & 0xFFFFu) << 16);                // tensor_dim0[15:0]
        g1[2] = (int)((NELEM >> 16) | (1u << 16));             // td0[31:16] | tensor_dim1=1
        g1[3] = (int)(NELEM << 16);                            // tile_dim0 = 40960
        g1[4] = 1;                                             // tile_dim1 = 1
        g1[5] = (int)NELEM;                                    // tensor_dim0_stride
        g1[6] = (int)((NELEM & 0xFFFFu) << 16);                // tensor_dim1_stride lo
        g1[7] = 0;

        i32x4 z4 = {0, 0, 0, 0};
        i32x8 z8 = {0, 0, 0, 0, 0, 0, 0, 0};
        __builtin_amdgcn_tensor_load_to_lds(g0, g1, z4, z4, z8, 0);
        __builtin_amdgcn_s_wait_tensorcnt(0);
    }

    // ---- warm the 1 MB packed RAM into L2 while the TDM runs ----
    {
        const char* base = (const char*)packedT;
#pragma unroll
        for (int i = 0; i < 16; ++i)
            __builtin_prefetch(base + (size_t)(tid + i * THREADS) * 64, 0, 1);
    }

    int bit_k[NPT];

    // ---- step 0: initial RAM addressed by position bits of i=0 (all zero) ----
#pragma unroll
    for (int k = 0; k < NPT; ++k) {
        int n = k * THREADS + tid;
        uint32_t addr = 0;
        const uint32_t p0 = 0u;                // 4 position bits of i=0
#pragma unroll
        for (int m = 0; m < NB_I; ++m) {
            uint32_t c = (uint32_t)init_conn[n * NB_I + m];   // in [0,4)
            addr = (addr << 1) | ((p0 >> (c & 3u)) & 1u);
        }
        uint32_t b = ((uint32_t)packedI[n] >> addr) & 1u;
        bit_k[k] = (int)b;
        __builtin_nontemporal_store(b ? 1.0f : 0.0f, &out[n]);
    }

    // ---- pack step-0 state into buffer 0 (+ position word for i=1) ----
#pragma unroll
    for (int k = 0; k < NPT; ++k) {
        uint32_t m = (uint32_t)__ballot(bit_k[k]);
        if (lane == 0) sbuf[k * 32 + wave] = m;
    }
    if (tid == 0) {
        uint32_t pw = 0;
        for (int j = 0; j < 4; ++j) pw |= (uint32_t)((1 >> (3 - j)) & 1) << j;
        sbuf[256] = pw;                        // input bits 8192..8195 for i=1
    }
    __syncthreads();                           // also publishes TDM'd connP

    // ---- main sequential loop: read buf[(t-1)&1], write buf[t&1] ----
    for (int t = 1; t < length; ++t) {
        const uint32_t* rb = sbuf + ((t - 1) & 1) * ST_STRIDE;
        uint32_t*       wb = sbuf + (t & 1) * ST_STRIDE;
        float* orow = out + (size_t)t * BITS;
#pragma unroll
        for (int k = 0; k < NPT; ++k) {
            int n = k * THREADS + tid;
            uint32_t addr = 0;
#pragma unroll
            for (int p = 0; p < NB_P; ++p) {
                uint32_t cc = connP[p * BITS + n];   // conflict-free ds_load_b32
                uint32_t c0 = cc & 0xFFFFu;          // conn[2p]   (MSB side)
                uint32_t c1 = cc >> 16;              // conn[2p+1]
                uint32_t b0 = (rb[c0 >> 5] >> (c0 & 31u)) & 1u;
                uint32_t b1 = (rb[c1 >> 5] >> (c1 & 31u)) & 1u;
                addr = (addr << 2) | (b0 << 1) | b1;
            }
            uint32_t word = packedT[(size_t)n * TM_WORDS + (addr >> 5)]; // L2 hit
            uint32_t b = (word >> (addr & 31u)) & 1u;
            bit_k[k] = (int)b;
            __builtin_nontemporal_store(b ? 1.0f : 0.0f, &orow[n]);
        }
        // write new state into the OTHER buffer: no barrier needed before this
#pragma unroll
        for (int k = 0; k < NPT; ++k) {
            uint32_t m = (uint32_t)__ballot(bit_k[k]);
            if (lane == 0) wb[k * 32 + wave] = m;     // one ds_store_b32 / wave
        }
        if (tid == 0) {
            int i = t + 1;
            uint32_t pw = 0;
            for (int j = 0; j < 4; ++j) pw |= (uint32_t)((i >> (3 - j)) & 1) << j;
            wb[256] = pw;
        }
        __syncthreads();                       // single barrier per step
    }
}

// ------------------------------- launch ------------------------------------

extern "C" void kernel_launch(void* const* d_in, const int* in_sizes, int n_in,
                              void* d_out, int out_size, void* d_ws, size_t ws_size,
                              hipStream_t stream) {
    (void)in_sizes; (void)n_in; (void)ws_size;

    const float* tm = (const float*)d_in[0];   // transition_memory [8192][1024]
    const float* im = (const float*)d_in[1];   // initial_memory    [8192][16]
    const int*   tc = (const int*)d_in[2];     // transition_connections [8192][10]
    const int*   ic = (const int*)d_in[3];     // initial_connections    [8192][4]
    float*      out = (float*)d_out;

    int length = out_size / BITS;              // host-side: can't read device scalar in capture

    uint32_t* connP = (uint32_t*)((char*)d_ws + WS_CONN_OFF);
    uint16_t* pI    = (uint16_t*)((char*)d_ws + WS_PI_OFF);
    uint32_t* pT    = (uint32_t*)((char*)d_ws + WS_PT_OFF);

    pack_transition_kernel<<<(BITS * TM_WORDS + 255) / 256, 256, 0, stream>>>(tm, pT);
    pack_initial_kernel<<<(BITS + 255) / 256, 256, 0, stream>>>(im, pI);
    pack_conn_kernel<<<(BITS * NB_P + 255) / 256, 256, 0, stream>>>(tc, connP);

    // 160 KB dynamic LDS (fits in the 320 KB WGP LDS); raise the per-kernel cap.
    (void)hipFuncSetAttribute((const void*)autoreg_ram_seq_kernel,
                              hipFuncAttributeMaxDynamicSharedMemorySize,
                              (int)SMEM_BYTES);

    autoreg_ram_seq_kernel<<<1, THREADS, SMEM_BYTES, stream>>>(
        connP, pT, pI, ic, out, length);
}